// EntmaxNsect_39427799777439
// MI455X (gfx1250) — compile-verified
//
#include <hip/hip_runtime.h>
#include <math.h>

// Entmax-1.5 via 5x5 n-section search, X: (rows, 32000) f32.
// alpha = 1.5  ->  Xs = 0.5*X,  p = relu(Xs - tau)^2,  hi offset = (1/d)^0.5.

#define ROW_D   32000
#define BLOCK_T 800                     // 25 wave32s
#define NWAVES  (BLOCK_T / 32)          // 25
#define V4PT    (ROW_D / 4 / BLOCK_T)   // 10 float4 per thread
#define VPT     (V4PT * 4)              // 40 floats per thread (register resident)

typedef float v4f __attribute__((ext_vector_type(4)));

__device__ __forceinline__ float wave_red_max(float v) {
#pragma unroll
  for (int o = 16; o > 0; o >>= 1) v = fmaxf(v, __shfl_xor(v, o, 32));
  return v;
}
__device__ __forceinline__ float wave_red_sum(float v) {
#pragma unroll
  for (int o = 16; o > 0; o >>= 1) v += __shfl_xor(v, o, 32);
  return v;
}

__global__ __launch_bounds__(BLOCK_T) void entmax15_nsect_kernel(
    const float* __restrict__ X, float* __restrict__ Out) {
  __shared__ float s_part[NWAVES * 4];
  __shared__ float s_res[4];

  const int tid  = threadIdx.x;
  const int lane = tid & 31;
  const int wid  = tid >> 5;
  const long long row = blockIdx.x;

  const v4f* __restrict__ xrow = (const v4f*)(X + row * (long long)ROW_D);
  v4f* __restrict__ orow = (v4f*)(Out + row * (long long)ROW_D);

  // ---- load the entire row into registers: X is read from HBM exactly once ----
  float x[VPT];
#pragma unroll
  for (int i = 0; i < V4PT; ++i) {
    v4f v = __builtin_nontemporal_load(&xrow[tid + i * BLOCK_T]);
    x[4 * i + 0] = 0.5f * v.x;          // Xs = (alpha-1) * X
    x[4 * i + 1] = 0.5f * v.y;
    x[4 * i + 2] = 0.5f * v.z;
    x[4 * i + 3] = 0.5f * v.w;
  }

  // ---- row max (thread-local max kept for the skip test below) ----
  float lmax = -INFINITY;
#pragma unroll
  for (int i = 0; i < VPT; ++i) lmax = fmaxf(lmax, x[i]);

  float wm = wave_red_max(lmax);
  if (lane == 0) s_part[wid] = wm;
  __syncthreads();
  if (wid == 0) {
    float t = (lane < NWAVES) ? s_part[lane] : -INFINITY;
    t = wave_red_max(t);
    if (lane == 0) s_res[0] = t;
  }
  __syncthreads();
  const float rmax = s_res[0];

  float lo = rmax - 1.0f;
  float hi = rmax - 0.0055901699f;      // (1/32000)^(alpha-1) = 1/sqrt(32000)

  // ---- 5 rounds of 5-section search; 4 interior points per register pass ----
#pragma unroll
  for (int it = 0; it < 5; ++it) {
    const float width = (hi - lo) * 0.2f;
    const float t1 = lo + width;
    const float t2 = lo + 2.0f * width;
    const float t3 = lo + 3.0f * width;
    const float t4 = lo + 4.0f * width;

    float s1 = 0.f, s2 = 0.f, s3 = 0.f, s4 = 0.f;
    if (lmax > t1) {                    // all relu terms provably zero otherwise
#pragma unroll
      for (int i = 0; i < VPT; ++i) {
        float d1 = fmaxf(x[i] - t1, 0.f); s1 = fmaf(d1, d1, s1);
        float d2 = fmaxf(x[i] - t2, 0.f); s2 = fmaf(d2, d2, s2);
        float d3 = fmaxf(x[i] - t3, 0.f); s3 = fmaf(d3, d3, s3);
        float d4 = fmaxf(x[i] - t4, 0.f); s4 = fmaf(d4, d4, s4);
      }
    }
    s1 = wave_red_sum(s1); s2 = wave_red_sum(s2);
    s3 = wave_red_sum(s3); s4 = wave_red_sum(s4);
    if (lane == 0) {
      s_part[wid * 4 + 0] = s1; s_part[wid * 4 + 1] = s2;
      s_part[wid * 4 + 2] = s3; s_part[wid * 4 + 3] = s4;
    }
    __syncthreads();
    if (wid == 0) {
      float r1 = (lane < NWAVES) ? s_part[lane * 4 + 0] : 0.f;
      float r2 = (lane < NWAVES) ? s_part[lane * 4 + 1] : 0.f;
      float r3 = (lane < NWAVES) ? s_part[lane * 4 + 2] : 0.f;
      float r4 = (lane < NWAVES) ? s_part[lane * 4 + 3] : 0.f;
      r1 = wave_red_sum(r1); r2 = wave_red_sum(r2);
      r3 = wave_red_sum(r3); r4 = wave_red_sum(r4);
      if (lane == 0) {
        s_res[0] = r1; s_res[1] = r2; s_res[2] = r3; s_res[3] = r4;
      }
    }
    __syncthreads();
    const float f1 = s_res[0], f2 = s_res[1], f3 = s_res[2], f4 = s_res[3];

    // sequential selection, identical to reference ordering: f(t) = sum - 1 >= 0
    float nlo = lo;
    if (f1 >= 1.0f) nlo = t1;
    if (f2 >= 1.0f) nlo = t2;
    if (f3 >= 1.0f) nlo = t3;
    if (f4 >= 1.0f) nlo = t4;
    lo = nlo;
    hi = nlo + width;
  }

  const float tau = 0.5f * (lo + hi);

  // ---- final p = relu(Xs - tau)^2 (reuse registers), and its row sum ----
  float lsum = 0.f;
#pragma unroll
  for (int i = 0; i < VPT; ++i) {
    float d = fmaxf(x[i] - tau, 0.f);
    d = d * d;
    x[i] = d;
    lsum += d;
  }
  float ws = wave_red_sum(lsum);
  if (lane == 0) s_part[wid] = ws;
  __syncthreads();
  if (wid == 0) {
    float t = (lane < NWAVES) ? s_part[lane] : 0.f;
    t = wave_red_sum(t);
    if (lane == 0) s_res[0] = t;
  }
  __syncthreads();
  const float inv = 1.0f / s_res[0];

  // ---- normalize and stream out with non-temporal B128 stores ----
#pragma unroll
  for (int i = 0; i < V4PT; ++i) {
    v4f v;
    v.x = x[4 * i + 0] * inv;
    v.y = x[4 * i + 1] * inv;
    v.z = x[4 * i + 2] * inv;
    v.w = x[4 * i + 3] * inv;
    __builtin_nontemporal_store(v, &orow[tid + i * BLOCK_T]);
  }
}

extern "C" void kernel_launch(void* const* d_in, const int* in_sizes, int n_in,
                              void* d_out, int out_size, void* d_ws, size_t ws_size,
                              hipStream_t stream) {
  (void)n_in; (void)out_size; (void)d_ws; (void)ws_size;
  const float* X = (const float*)d_in[0];
  float* Out = (float*)d_out;
  const int rows = in_sizes[0] / ROW_D;   // 4096 for the reference shape
  entmax15_nsect_kernel<<<rows, BLOCK_T, 0, stream>>>(X, Out);
}